// ImprovedGATv2Layer_29686813950822
// MI455X (gfx1250) — compile-verified
//
#include <hip/hip_runtime.h>
#include <math.h>

#define N_NODES 50000
#define E_EDGES 800000
#define IN_DIM  128
#define HEADS   4
#define OUT_DIM 32
#define TOT     128
#define E_TOT   (E_EDGES + N_NODES)   // 850000 (self loops appended)
#define LN_EPS  1e-5f
#define NEG_SLOPE 0.2f

typedef float v2f __attribute__((ext_vector_type(2)));
typedef float v8f __attribute__((ext_vector_type(8)));

// ---- order-preserving float <-> uint encoding for atomic max on f32 ----
__device__ __forceinline__ unsigned fenc(float f) {
    unsigned u = __float_as_uint(f);
    return (u & 0x80000000u) ? ~u : (u | 0x80000000u);
}
__device__ __forceinline__ float fdec(unsigned u) {
    unsigned v = (u & 0x80000000u) ? (u & 0x7FFFFFFFu) : ~u;
    return __uint_as_float(v);
}
__device__ __forceinline__ float lrelu(float v) {
    return v > 0.0f ? v : NEG_SLOPE * v;
}

// ============================================================================
// K0: zero the output accumulator, init segment-max (enc(-inf)-safe 0) & denom
// ============================================================================
__global__ void gat_init_kernel(float* __restrict__ out,
                                unsigned* __restrict__ smax,
                                float* __restrict__ denom) {
    long long tid = (long long)blockIdx.x * blockDim.x + threadIdx.x;
    if (tid < (long long)N_NODES * TOT) out[tid] = 0.0f;
    if (tid < (long long)N_NODES * HEADS) {
        smax[tid]  = 0u;     // every dst has a self loop -> always overwritten
        denom[tid] = 0.0f;
    }
}

// ============================================================================
// K1: dual GEMM  xl = x@W_l + b_l ; xr = x@W_r + b_r   via V_WMMA_F32_16X16X4
// block = 512 threads = 16 waves; wave w: mat = w/8 (l/r), ntile = w%8 (N cols)
// A (16x4 f32): lane<16 rows, VGPR0=K{0|2}, VGPR1=K{1|3} by half-wave
// B (4x16 f32): mirrored; C/D (16x16 f32): VGPR v -> rows v / v+8, N=lane%16
// ============================================================================
__global__ __launch_bounds__(512)
void gat_gemm_kernel(const float* __restrict__ x,
                     const float* __restrict__ Wl, const float* __restrict__ bl,
                     const float* __restrict__ Wr, const float* __restrict__ br,
                     float* __restrict__ xl, float* __restrict__ xr) {
    __shared__ float xs[16 * IN_DIM];               // 8 KB slab of 16 node rows
    const int tid  = threadIdx.x;
    const int row0 = blockIdx.x * 16;

    // stage x[row0:row0+16, :] -> LDS (512 threads x 1 float4)
    ((float4*)xs)[tid] = ((const float4*)(x + (size_t)row0 * IN_DIM))[tid];
    __syncthreads();

    const int wave  = tid >> 5;
    const int lane  = tid & 31;
    const int half  = lane >> 4;       // 0: lanes 0-15, 1: lanes 16-31
    const int lpos  = lane & 15;
    const int mat   = wave >> 3;       // 0 -> W_l/xl, 1 -> W_r/xr
    const int ntile = wave & 7;
    const int col   = ntile * 16 + lpos;

    const float* __restrict__ W    = mat ? Wr : Wl;
    const float* __restrict__ bvec = mat ? br : bl;
    float*       __restrict__ outm = mat ? xr : xl;

    v8f c = {};
#pragma unroll
    for (int k = 0; k < IN_DIM; k += 4) {
        const int kk = k + 2 * half;
        v2f a, b;
        a[0] = xs[lpos * IN_DIM + kk];
        a[1] = xs[lpos * IN_DIM + kk + 1];
        b[0] = W[(size_t)kk * TOT + col];
        b[1] = W[(size_t)(kk + 1) * TOT + col];
        c = __builtin_amdgcn_wmma_f32_16x16x4_f32(
                /*neg_a=*/false, a, /*neg_b=*/false, b,
                /*c_mod=*/(short)0, c, /*reuse_a=*/false, /*reuse_b=*/false);
    }

    const float bv = bvec[col];
#pragma unroll
    for (int v = 0; v < 8; ++v)
        outm[(size_t)(row0 + v + 8 * half) * TOT + col] = c[v] + bv;
}

// ============================================================================
// K2: per-(edge,head) attention logit + segment max over dst
// ============================================================================
__global__ void gat_score_kernel(const int* __restrict__ ei,
                                 const float* __restrict__ xl,
                                 const float* __restrict__ xr,
                                 const float* __restrict__ att,
                                 float* __restrict__ sbuf,
                                 unsigned* __restrict__ smax) {
    long long tid = (long long)blockIdx.x * blockDim.x + threadIdx.x;
    if (tid >= (long long)E_TOT * HEADS) return;
    const int e = (int)(tid >> 2);
    const int h = (int)(tid & 3);
    int src, dst;
    if (e < E_EDGES) { src = ei[e]; dst = ei[E_EDGES + e]; }
    else             { src = dst = e - E_EDGES; }

    const float4* pl = (const float4*)(xl + (size_t)src * TOT + h * OUT_DIM);
    const float4* pr = (const float4*)(xr + (size_t)dst * TOT + h * OUT_DIM);
    const float4* pa = (const float4*)(att + h * OUT_DIM);

    float s = 0.0f;
#pragma unroll
    for (int q = 0; q < OUT_DIM / 4; ++q) {
        float4 a = pl[q], b = pr[q], w = pa[q];
        s += lrelu(a.x + b.x) * w.x;
        s += lrelu(a.y + b.y) * w.y;
        s += lrelu(a.z + b.z) * w.z;
        s += lrelu(a.w + b.w) * w.w;
    }
    sbuf[(size_t)e * HEADS + h] = s;
    atomicMax(&smax[(size_t)dst * HEADS + h], fenc(s));
}

// ============================================================================
// K3: e = exp(s - smax[dst]); segment-sum denominator
// ============================================================================
__global__ void gat_expsum_kernel(const int* __restrict__ ei,
                                  const unsigned* __restrict__ smax,
                                  float* __restrict__ sbuf,
                                  float* __restrict__ denom) {
    long long tid = (long long)blockIdx.x * blockDim.x + threadIdx.x;
    if (tid >= (long long)E_TOT * HEADS) return;
    const int e = (int)(tid >> 2);
    const int h = (int)(tid & 3);
    const int dst = (e < E_EDGES) ? ei[E_EDGES + e] : (e - E_EDGES);
    const float m  = fdec(smax[(size_t)dst * HEADS + h]);
    const float ev = expf(sbuf[(size_t)e * HEADS + h] - m);
    sbuf[(size_t)e * HEADS + h] = ev;
    atomicAdd(&denom[(size_t)dst * HEADS + h], ev);
}

// ============================================================================
// K4: out[dst] += alpha * xl[src]   (thread = edge x head x 8-feature chunk)
// ============================================================================
__global__ void gat_scatter_kernel(const int* __restrict__ ei,
                                   const float* __restrict__ xl,
                                   const float* __restrict__ sbuf,
                                   const float* __restrict__ denom,
                                   float* __restrict__ out) {
    long long tid = (long long)blockIdx.x * blockDim.x + threadIdx.x;
    if (tid >= (long long)E_TOT * HEADS * 4) return;
    const int e = (int)(tid >> 4);
    const int h = (int)((tid >> 2) & 3);
    const int q = (int)(tid & 3);
    int src, dst;
    if (e < E_EDGES) { src = ei[e]; dst = ei[E_EDGES + e]; }
    else             { src = dst = e - E_EDGES; }

    const float alpha = sbuf[(size_t)e * HEADS + h] /
                        denom[(size_t)dst * HEADS + h];
    const size_t fo = (size_t)h * OUT_DIM + q * 8;
    const float4* p = (const float4*)(xl + (size_t)src * TOT + fo);
    float* o = out + (size_t)dst * TOT + fo;
    float4 a = p[0], b = p[1];
    atomicAdd(&o[0], alpha * a.x);
    atomicAdd(&o[1], alpha * a.y);
    atomicAdd(&o[2], alpha * a.z);
    atomicAdd(&o[3], alpha * a.w);
    atomicAdd(&o[4], alpha * b.x);
    atomicAdd(&o[5], alpha * b.y);
    atomicAdd(&o[6], alpha * b.z);
    atomicAdd(&o[7], alpha * b.w);
}

// ============================================================================
// K5: out = ELU(LayerNorm(out + bias + x))  — one wave32 per node
// ============================================================================
__global__ __launch_bounds__(256)
void gat_final_kernel(const float* __restrict__ x,
                      const float* __restrict__ bias,
                      const float* __restrict__ ln_g,
                      const float* __restrict__ ln_b,
                      float* __restrict__ out) {
    const int wave = threadIdx.x >> 5;
    const int lane = threadIdx.x & 31;
    const int node = blockIdx.x * 8 + wave;
    if (node >= N_NODES) return;

    float v[4];
    float sum = 0.0f;
#pragma unroll
    for (int i = 0; i < 4; ++i) {
        const int idx = lane + i * 32;
        v[i] = out[(size_t)node * TOT + idx] + bias[idx] + x[(size_t)node * TOT + idx];
        sum += v[i];
    }
#pragma unroll
    for (int m = 16; m >= 1; m >>= 1) sum += __shfl_xor(sum, m, 32);
    const float mu = sum * (1.0f / TOT);

    float var = 0.0f;
#pragma unroll
    for (int i = 0; i < 4; ++i) { const float d = v[i] - mu; var += d * d; }
#pragma unroll
    for (int m = 16; m >= 1; m >>= 1) var += __shfl_xor(var, m, 32);
    var *= (1.0f / TOT);

    const float inv = 1.0f / sqrtf(var + LN_EPS);
#pragma unroll
    for (int i = 0; i < 4; ++i) {
        const int idx = lane + i * 32;
        float y = (v[i] - mu) * inv * ln_g[idx] + ln_b[idx];
        y = y > 0.0f ? y : (expf(y) - 1.0f);
        out[(size_t)node * TOT + idx] = y;
    }
}

// ============================================================================
extern "C" void kernel_launch(void* const* d_in, const int* in_sizes, int n_in,
                              void* d_out, int out_size, void* d_ws, size_t ws_size,
                              hipStream_t stream) {
    const float* x    = (const float*)d_in[0];
    const int*   ei   = (const int*)  d_in[1];
    const float* Wl   = (const float*)d_in[2];
    const float* bl   = (const float*)d_in[3];
    const float* Wr   = (const float*)d_in[4];
    const float* br   = (const float*)d_in[5];
    const float* att  = (const float*)d_in[6];
    const float* bias = (const float*)d_in[7];
    const float* lng  = (const float*)d_in[8];
    const float* lnb  = (const float*)d_in[9];
    float* out = (float*)d_out;

    // workspace layout
    float*    xl    = (float*)d_ws;
    float*    xr    = xl + (size_t)N_NODES * TOT;
    float*    sbuf  = xr + (size_t)N_NODES * TOT;
    unsigned* smax  = (unsigned*)(sbuf + (size_t)E_TOT * HEADS);
    float*    denom = (float*)(smax + (size_t)N_NODES * HEADS);

    const long long nInit = (long long)N_NODES * TOT;
    gat_init_kernel<<<(int)((nInit + 255) / 256), 256, 0, stream>>>(out, smax, denom);

    gat_gemm_kernel<<<N_NODES / 16, 512, 0, stream>>>(x, Wl, bl, Wr, br, xl, xr);

    const long long nEH = (long long)E_TOT * HEADS;
    gat_score_kernel<<<(int)((nEH + 255) / 256), 256, 0, stream>>>(ei, xl, xr, att, sbuf, smax);
    gat_expsum_kernel<<<(int)((nEH + 255) / 256), 256, 0, stream>>>(ei, smax, sbuf, denom);

    const long long nScat = nEH * 4;
    gat_scatter_kernel<<<(int)((nScat + 255) / 256), 256, 0, stream>>>(ei, xl, sbuf, denom, out);

    gat_final_kernel<<<(N_NODES + 7) / 8, 256, 0, stream>>>(x, bias, lng, lnb, out);
}